// PositionEmbeddingSineOnlineSegment_81131932221734
// MI455X (gfx1250) — compile-verified
//
#include <hip/hip_runtime.h>
#include <math.h>

#define BSZ      32
#define LEN      8192
#define SEG      512
#define NF       256
#define THREADS1 1024
#define PER_T    (LEN / THREADS1)   // 8 mask elements per thread
#define THREADS2 256
#define ITER     8                  // positions-per-thread loop in pass 2

typedef float v4f __attribute__((ext_vector_type(4)));

// ---------------------------------------------------------------------------
// Pass 1: one workgroup per batch row. Block-wide scan of the 0/1 mask in LDS,
// first-valid index via shared atomicMin, then emit normalized segmented
// x_embed (one float per position) into workspace. 2 MB total traffic.
// ---------------------------------------------------------------------------
__global__ __launch_bounds__(THREADS1)
void pe_scan_kernel(const int* __restrict__ mask, float* __restrict__ xe) {
  __shared__ int sC[LEN];        // inclusive cumsum of the row (exact, int)
  __shared__ int sSum[THREADS1]; // per-thread totals for the block scan
  __shared__ int sStart;

  const int b   = blockIdx.x;
  const int tid = threadIdx.x;
  const int* row = mask + (size_t)b * LEN;
  const int base = tid * PER_T;

  if (tid == 0) sStart = 0x7FFFFFFF;
  // gfx1250 prefetch path: pull the mask row toward the WGP
  __builtin_prefetch(row + base, 0, 1);
  __syncthreads();

  // load 8 consecutive ints (two b128 loads)
  int4 a0 = ((const int4*)(row + base))[0];
  int4 a1 = ((const int4*)(row + base))[1];
  int v[PER_T] = {a0.x, a0.y, a0.z, a0.w, a1.x, a1.y, a1.z, a1.w};

  int run = 0, incl[PER_T];
  int firstLocal = 0x7FFFFFFF;
#pragma unroll
  for (int i = 0; i < PER_T; ++i) {
    const int m = (v[i] == 1) ? 1 : 0;
    if (m && firstLocal == 0x7FFFFFFF) firstLocal = base + i;
    run += m;
    incl[i] = run;
  }
  sSum[tid] = run;
  if (firstLocal != 0x7FFFFFFF) atomicMin(&sStart, firstLocal);
  __syncthreads();

  // Hillis-Steele inclusive scan over the 1024 per-thread totals
#pragma unroll
  for (int off = 1; off < THREADS1; off <<= 1) {
    const int t = (tid >= off) ? sSum[tid - off] : 0;
    __syncthreads();
    sSum[tid] += t;
    __syncthreads();
  }
  const int excl = sSum[tid] - run;
#pragma unroll
  for (int i = 0; i < PER_T; ++i) sC[base + i] = excl + incl[i];
  const int startRaw = sStart;   // atomicMin writes happened before the scan barriers
  __syncthreads();

  const bool hv = (startRaw != 0x7FFFFFFF);
  const int  st = hv ? startRaw : 0;

#pragma unroll
  for (int i = 0; i < PER_T; ++i) {
    const int l   = base + i;
    const int rel = l - st;
    float out = 0.0f;
    if (hv && rel >= 0) {
      const int segId    = rel >> 9;                 // / SEG
      const int segStart = st + (segId << 9);
      const int segEnd   = min(segStart + SEG, LEN); // exclusive
      const int cPrev    = (segStart > 0) ? sC[segStart - 1] : 0;
      const float num = (float)(sC[l] - cPrev);
      const float den = (float)(sC[segEnd - 1] - cPrev) + 1e-6f;
      out = num / den * 6.28318530717958647692f;     // * 2*pi (SCALE)
    }
    xe[(size_t)b * LEN + l] = out;
  }
}

// ---------------------------------------------------------------------------
// Pass 2: streaming expansion. Each thread owns 4 consecutive features
// (one float4 = 2 sin + 2 cos), reuses its exp2f-derived inverse dim_t across
// ITER positions, and writes with non-temporal b128 stores (256 MB > 192 MB L2,
// so bypass cache rinse on the write-once output).
// out[p, 4t+0] = sin(x*inv0), out[p, 4t+1] = cos(x*inv0)
// out[p, 4t+2] = sin(x*inv1), out[p, 4t+3] = cos(x*inv1)
// ---------------------------------------------------------------------------
__global__ __launch_bounds__(THREADS2)
void pe_sincos_kernel(const float* __restrict__ xe, float* __restrict__ out) {
  const int tid   = threadIdx.x;
  const int tFeat = tid & 63;   // float4 slot within the 256-feature row
  const int tPos  = tid >> 6;   // 0..3 positions in parallel per block

  const float k    = 0.051905126482615f;  // log2(10000) / 256
  const float inv0 = exp2f(-(float)(4 * tFeat)     * k);  // 1/dim_t, pair 0
  const float inv1 = exp2f(-(float)(4 * tFeat + 2) * k);  // 1/dim_t, pair 1

  int p = blockIdx.x * (4 * ITER) + tPos;
#pragma unroll
  for (int i = 0; i < ITER; ++i, p += 4) {
    const float x  = xe[p];           // wave-uniform, L1-broadcast
    const float a0 = x * inv0;
    const float a1 = x * inv1;        // args in [0, 2*pi]: v_sin/v_cos exact enough
    v4f o;
    o.x = __sinf(a0);
    o.y = __cosf(a0);
    o.z = __sinf(a1);
    o.w = __cosf(a1);
    __builtin_nontemporal_store(o, (v4f*)out + (size_t)p * (NF / 4) + tFeat);
  }
}

extern "C" void kernel_launch(void* const* d_in, const int* in_sizes, int n_in,
                              void* d_out, int out_size, void* d_ws, size_t ws_size,
                              hipStream_t stream) {
  // setup_inputs order: d_in[0] = x (float32, unused by the reference math),
  //                     d_in[1] = mask (int32, BSZ x LEN)
  const int* mask = (const int*)d_in[1];
  float* xe  = (float*)d_ws;   // BSZ*LEN floats = 1 MB scratch
  float* out = (float*)d_out;  // BSZ*LEN*NF floats

  pe_scan_kernel<<<BSZ, THREADS1, 0, stream>>>(mask, xe);

  const int nPos   = BSZ * LEN;            // 262144
  const int blocks = nPos / (4 * ITER);    // 8192 blocks * 256 threads
  pe_sincos_kernel<<<blocks, THREADS2, 0, stream>>>(xe, out);
}